// MedGCNRelationAttention_45827301048842
// MI455X (gfx1250) — compile-verified
//
#include <hip/hip_runtime.h>
#include <math.h>

#define NN 50000
#define RR 4
#define EE 1600000
#define HH 256
#define CC 16

typedef __attribute__((ext_vector_type(2))) float v2f;
typedef __attribute__((ext_vector_type(8))) float v8f;

// ---------------- degree / norm ----------------
__global__ void deg_kernel(const int* __restrict__ ei,
                           float* __restrict__ deg_r, float* __restrict__ deg_c) {
    long tid = (long)blockIdx.x * blockDim.x + threadIdx.x;
    if (tid >= (long)RR * EE) return;
    int r = (int)(tid / EE);
    int e = (int)(tid % EE);
    int row = ei[(size_t)r * 2 * EE + e];
    int col = ei[(size_t)r * 2 * EE + EE + e];
    atomicAdd(&deg_r[r * NN + row], 1.0f);
    atomicAdd(&deg_c[r * NN + col], 1.0f);
}

__global__ void rsqrt_kernel(float* __restrict__ d, int n) {
    int i = blockIdx.x * blockDim.x + threadIdx.x;
    if (i < n) d[i] = rsqrtf(fmaxf(d[i], 1.0f));
}

// ---------------- SPMM: one wave32 per edge ----------------
__global__ void __launch_bounds__(256)
spmm_kernel(const float* __restrict__ feat, float* __restrict__ S,
            const int* __restrict__ ei,
            const float* __restrict__ dinv_r, const float* __restrict__ dinv_c,
            int r) {
    int wave = threadIdx.x >> 5;
    int lane = threadIdx.x & 31;
    long e = (long)blockIdx.x * 8 + wave;
    if (e >= EE) return;
    int row = ei[(size_t)r * 2 * EE + e];
    int col = ei[(size_t)r * 2 * EE + EE + e];
    float w = dinv_r[r * NN + row] * dinv_c[r * NN + col];
    const float4* src = (const float4*)(feat + (size_t)col * HH);
    float* dst = S + (size_t)row * HH;
    float4 v0 = src[lane];          // floats [lane*4 .. lane*4+3]   (0..127)
    float4 v1 = src[lane + 32];     // floats [128+lane*4 .. +3]     (128..255)
    atomicAdd(dst + lane * 4 + 0, w * v0.x);
    atomicAdd(dst + lane * 4 + 1, w * v0.y);
    atomicAdd(dst + lane * 4 + 2, w * v0.z);
    atomicAdd(dst + lane * 4 + 3, w * v0.w);
    atomicAdd(dst + 128 + lane * 4 + 0, w * v1.x);
    atomicAdd(dst + 128 + lane * 4 + 1, w * v1.y);
    atomicAdd(dst + 128 + lane * 4 + 2, w * v1.z);
    atomicAdd(dst + 128 + lane * 4 + 3, w * v1.w);
}

// ---------------- GEMM + ReLU via V_WMMA_F32_16X16X4_F32 ----------------
// out[16x256 tile] = relu(S[16x256] @ W[256x256]); optionally out += scale*relu(...)
__global__ void __launch_bounds__(256)
gemm_relu_kernel(const float* __restrict__ S, const float* __restrict__ W,
                 float* __restrict__ out, float scale, int accumulate) {
    __shared__ float As[16 * 257];               // padded stride vs 64 banks
    int mbase = blockIdx.x * 16;

    // cooperative load of the 16x256 A strip (1024 float4s, 4 per thread)
    for (int i = 0; i < 4; ++i) {
        int idx = threadIdx.x + i * 256;         // float4 index 0..1023
        int row = idx >> 6;                      // /64
        int c4  = idx & 63;
        float4 v = *(const float4*)(S + (size_t)(mbase + row) * HH + c4 * 4);
        float* dstp = &As[row * 257 + c4 * 4];
        dstp[0] = v.x; dstp[1] = v.y; dstp[2] = v.z; dstp[3] = v.w;
    }
    __syncthreads();

    int wave = threadIdx.x >> 5;
    int lane = threadIdx.x & 31;
    bool hi  = lane >= 16;
    int  l   = lane & 15;
    int  n0  = wave * 32;                        // each wave: two 16-wide N tiles
    int  n1  = n0 + 16;

    v8f c0 = {}; v8f c1 = {};
    for (int k = 0; k < HH; k += 4) {
        if (k + 4 < HH) {                        // gfx1250 global_prefetch_b8
            __builtin_prefetch(&W[(size_t)(k + 4) * HH + n0 + l], 0, 1);
        }
        // A fragment (16x4 f32): lanes 0-15 -> K = k,k+1 ; lanes 16-31 -> K = k+2,k+3
        int ko = hi ? k + 2 : k;
        v2f a;
        a.x = As[l * 257 + ko];
        a.y = As[l * 257 + ko + 1];
        // B fragment (4x16 f32): same K split, N = n0+l (resp. n1+l)
        v2f b0, b1;
        b0.x = W[(size_t)ko * HH + n0 + l];
        b0.y = W[(size_t)(ko + 1) * HH + n0 + l];
        b1.x = W[(size_t)ko * HH + n1 + l];
        b1.y = W[(size_t)(ko + 1) * HH + n1 + l];
        c0 = __builtin_amdgcn_wmma_f32_16x16x4_f32(false, a, false, b0, (short)0, c0, false, false);
        c1 = __builtin_amdgcn_wmma_f32_16x16x4_f32(false, a, false, b1, (short)0, c1, false, false);
    }

    // D layout: VGPR v holds M=v (lanes 0-15) and M=v+8 (lanes 16-31); N = lane%16
    int mrow = mbase + (hi ? 8 : 0);
    for (int v = 0; v < 8; ++v) {
        float x0 = fmaxf(c0[v], 0.0f);
        float x1 = fmaxf(c1[v], 0.0f);
        size_t i0 = (size_t)(mrow + v) * HH + n0 + l;
        size_t i1 = (size_t)(mrow + v) * HH + n1 + l;
        if (accumulate) { out[i0] += scale * x0; out[i1] += scale * x1; }
        else            { out[i0] = x0;          out[i1] = x1; }
    }
}

// ---------------- attention + residual fusion + output projection ----------------
__global__ void __launch_bounds__(256)
attn_out_kernel(const float* __restrict__ H2, const float* __restrict__ att_q,
                const float* __restrict__ tau_p, const float* __restrict__ W_out,
                const float* __restrict__ b_out,
                float* __restrict__ logits, float* __restrict__ alpha_out) {
    __shared__ float h2s[8][HH];
    int wave = threadIdx.x >> 5;
    int lane = threadIdx.x & 31;
    int node = blockIdx.x * 8 + wave;            // NN/8 blocks, exact

    float q[8];
    for (int i = 0; i < 8; ++i) q[i] = att_q[lane * 8 + i];

    float chunk[RR][8];
    float s[RR];
    for (int r = 0; r < RR; ++r) {
        const float* base = H2 + ((size_t)r * NN + node) * HH + lane * 8;
        float4 a = *(const float4*)(base);
        float4 b = *(const float4*)(base + 4);
        chunk[r][0] = a.x; chunk[r][1] = a.y; chunk[r][2] = a.z; chunk[r][3] = a.w;
        chunk[r][4] = b.x; chunk[r][5] = b.y; chunk[r][6] = b.z; chunk[r][7] = b.w;
        float p = 0.0f;
        for (int i = 0; i < 8; ++i) p += chunk[r][i] * q[i];
        for (int off = 16; off >= 1; off >>= 1) p += __shfl_xor(p, off, 32);
        s[r] = p;
    }

    float tau_c = fminf(fmaxf(tau_p[0], 0.5f), 5.0f);
    float t[RR], m = -3.4e38f;
    for (int r = 0; r < RR; ++r) { t[r] = s[r] / tau_c; m = fmaxf(m, t[r]); }
    float al[RR], sum = 0.0f;
    for (int r = 0; r < RR; ++r) { al[r] = __expf(t[r] - m); sum += al[r]; }
    float inv = 1.0f / sum;
    for (int r = 0; r < RR; ++r) al[r] *= inv;

    // h2 = mean_r H2 + sum_r alpha_r H2  ->  coefficient (0.25 + alpha_r)
    for (int i = 0; i < 8; ++i) {
        float h = 0.0f;
        for (int r = 0; r < RR; ++r) h += (0.25f + al[r]) * chunk[r][i];
        h2s[wave][lane * 8 + i] = h;
    }
    __syncthreads();

    if (lane < CC) {
        float acc = b_out[lane];
        for (int h = 0; h < HH; ++h) acc += h2s[wave][h] * W_out[h * CC + lane];
        logits[(size_t)node * CC + lane] = acc;
    }
    if (lane < RR) alpha_out[(size_t)node * RR + lane] = al[lane];
}

// ---------------- driver ----------------
extern "C" void kernel_launch(void* const* d_in, const int* in_sizes, int n_in,
                              void* d_out, int out_size, void* d_ws, size_t ws_size,
                              hipStream_t stream) {
    (void)in_sizes; (void)n_in; (void)out_size; (void)ws_size;
    const float* X     = (const float*)d_in[0];
    const int*   ei    = (const int*)d_in[1];
    const float* W1    = (const float*)d_in[2];
    const float* W2    = (const float*)d_in[3];
    const float* att_q = (const float*)d_in[4];
    const float* tau   = (const float*)d_in[5];
    const float* W_out = (const float*)d_in[6];
    const float* b_out = (const float*)d_in[7];
    float* out = (float*)d_out;

    float* f      = (float*)d_ws;
    float* dinv_r = f;                                   // RR*NN
    float* dinv_c = dinv_r + (size_t)RR * NN;            // RR*NN
    float* S      = dinv_c + (size_t)RR * NN;            // NN*HH (per-relation scratch)
    float* h1     = S + (size_t)NN * HH;                 // NN*HH
    float* H2     = h1 + (size_t)NN * HH;                // RR*NN*HH

    // degrees (shared across both layers)
    hipMemsetAsync(dinv_r, 0, (size_t)2 * RR * NN * sizeof(float), stream);
    {
        long tot = (long)RR * EE;
        deg_kernel<<<(unsigned)((tot + 255) / 256), 256, 0, stream>>>(ei, dinv_r, dinv_c);
    }
    rsqrt_kernel<<<(2 * RR * NN + 255) / 256, 256, 0, stream>>>(dinv_r, 2 * RR * NN);

    // layer 1: h1 = mean_r relu((A_r X) W1_r)
    hipMemsetAsync(h1, 0, (size_t)NN * HH * sizeof(float), stream);
    for (int r = 0; r < RR; ++r) {
        hipMemsetAsync(S, 0, (size_t)NN * HH * sizeof(float), stream);
        spmm_kernel<<<EE / 8, 256, 0, stream>>>(X, S, ei, dinv_r, dinv_c, r);
        gemm_relu_kernel<<<NN / 16, 256, 0, stream>>>(S, W1 + (size_t)r * HH * HH, h1, 0.25f, 1);
    }

    // layer 2: H2[r] = relu((A_r h1) W2_r)
    for (int r = 0; r < RR; ++r) {
        hipMemsetAsync(S, 0, (size_t)NN * HH * sizeof(float), stream);
        spmm_kernel<<<EE / 8, 256, 0, stream>>>(h1, S, ei, dinv_r, dinv_c, r);
        gemm_relu_kernel<<<NN / 16, 256, 0, stream>>>(S, W2 + (size_t)r * HH * HH,
                                                      H2 + (size_t)r * NN * HH, 1.0f, 0);
    }

    // attention + fusion + projection; outputs: logits [N,16] then alpha [N,4]
    attn_out_kernel<<<NN / 8, 256, 0, stream>>>(H2, att_q, tau, W_out, b_out,
                                                out, out + (size_t)NN * CC);
}